// AtomicNumberPooling_12945031430717
// MI455X (gfx1250) — compile-verified
//
#include <hip/hip_runtime.h>
#include <stdint.h>

// AtomicNumberPooling for MI455X (gfx1250):
// segment-sum  out[b, (z-1)*F + f] += x[n, f]  over sorted `batch`.
// One block per graph; LDS accumulator (no global atomics); async
// global->LDS DMA staging (ASYNCcnt) double-buffered per wave.
// NT cache hints: input streamed once (async loads), output written once
// (nontemporal stores).

#define F          128
#define MAX_Z      100
#define TPB        256
#define NWAVES     (TPB / 32)
#define G          8                        // rows per async batch per wave
#define ROW_BYTES  (F * 4)                  // 512 B per node row
#define ACC_FLOATS (MAX_Z * F)              // 12800
#define ACC_BYTES  (ACC_FLOATS * 4)         // 51200
#define STAGE_BYTES (NWAVES * 2 * G * ROW_BYTES) // 65536
#define SMEM_BYTES (ACC_BYTES + STAGE_BYTES)

typedef float v4f __attribute__((ext_vector_type(4)));

__global__ __launch_bounds__(TPB, 1)
void anp_pool_kernel(const float* __restrict__ x,
                     const int*   __restrict__ z,
                     const int*   __restrict__ batch,
                     float*       __restrict__ out,
                     int N) {
  __shared__ __align__(16) unsigned char smem[SMEM_BYTES];
  __shared__ int sRange[2];
  float* acc = reinterpret_cast<float*>(smem);
  const int b   = blockIdx.x;
  const int tid = threadIdx.x;

  // Zero accumulator tile (each block writes its full output tile later,
  // so d_out never needs a separate zeroing pass).
  {
    v4f z4 = (v4f){0.f, 0.f, 0.f, 0.f};
    v4f* a4 = reinterpret_cast<v4f*>(smem);
    for (int i = tid; i < ACC_FLOATS / 4; i += TPB) a4[i] = z4;
  }

  // Thread 0: binary-search node range [s, e) of graph b (batch is sorted).
  if (tid == 0) {
    int lo = 0, hi = N;
    while (lo < hi) { int mid = (lo + hi) >> 1; if (batch[mid] < b) lo = mid + 1; else hi = mid; }
    sRange[0] = lo;
    hi = N;
    while (lo < hi) { int mid = (lo + hi) >> 1; if (batch[mid] < b + 1) lo = mid + 1; else hi = mid; }
    sRange[1] = lo;
  }
  __syncthreads();
  const int s = sRange[0], e = sRange[1];

  const int wid  = tid >> 5;
  const int lane = tid & 31;

  // Each wave owns a contiguous sub-range of this graph's rows.
  const int total   = e - s;
  const int per     = (total + NWAVES - 1) / NWAVES;
  const int myStart = s + wid * per;
  const int cnt     = min(per, e - myStart);   // may be <= 0

  // LDS byte address of this wave's staging area (low 32 bits of a generic
  // LDS pointer are the LDS byte offset on CDNA5).
  const uint32_t stageBase =
      (uint32_t)(uintptr_t)(smem + ACC_BYTES + wid * (2 * G * ROW_BYTES));

  if (cnt > 0) {
    const int nb = (cnt + G - 1) / G;

    // Prologue: issue async DMA for batch 0 (contiguous rows, 512B each;
    // 32 lanes x b128 = one full row per instruction). NT: streamed once.
    {
      const int m = min(G, cnt);
      uint64_t ga = (uint64_t)(uintptr_t)(x + (size_t)myStart * F) + (uint32_t)(lane * 16);
      uint32_t la = stageBase + (uint32_t)(lane * 16);
      for (int i = 0; i < m; ++i) {
        asm volatile("global_load_async_to_lds_b128 %0, %1, off th:TH_LOAD_NT"
                     :: "v"(la), "v"(ga) : "memory");
        ga += ROW_BYTES; la += ROW_BYTES;
      }
    }

    for (int k = 0; k < nb; ++k) {
      // Issue next batch into the other buffer (double-buffered).
      int mNext = 0;
      if (k + 1 < nb) {
        mNext = min(G, cnt - (k + 1) * G);
        uint64_t ga = (uint64_t)(uintptr_t)(x + (size_t)(myStart + (k + 1) * G) * F)
                      + (uint32_t)(lane * 16);
        uint32_t la = stageBase + (uint32_t)(((k + 1) & 1) * (G * ROW_BYTES) + lane * 16);
        for (int i = 0; i < mNext; ++i) {
          asm volatile("global_load_async_to_lds_b128 %0, %1, off th:TH_LOAD_NT"
                       :: "v"(la), "v"(ga) : "memory");
          ga += ROW_BYTES; la += ROW_BYTES;
        }
      }
      // Async loads complete in order: with G newer loads outstanding,
      // asynccnt <= G  ==>  all of batch k has landed in LDS.
      if (mNext == G) {
        asm volatile("s_wait_asynccnt %0" :: "i"(G) : "memory");
      } else {
        asm volatile("s_wait_asynccnt 0" ::: "memory");
      }

      const int m = min(G, cnt - k * G);
      const unsigned char* buf =
          smem + ACC_BYTES + wid * (2 * G * ROW_BYTES) + (k & 1) * (G * ROW_BYTES);
      for (int i = 0; i < m; ++i) {
        const int row = myStart + k * G + i;
        const int zr  = z[row];                      // uniform across the wave
        float* accRow = acc + (size_t)(zr - 1) * F;
        const float* srow = reinterpret_cast<const float*>(buf + i * ROW_BYTES);
        // lane l handles columns {l, l+32, l+64, l+96}: 32 distinct LDS banks
        // for both the ds_load_b32 reads and the ds_add_f32 atomics.
#pragma unroll
        for (int c = 0; c < 4; ++c) {
          const int col = lane + c * 32;
          atomicAdd(&accRow[col], srow[col]);
        }
      }
    }
  }

  __syncthreads();

  // Write the full 100x128 tile for this graph: coalesced b128 stores,
  // region exclusively owned by this block -> no global atomics.
  // Nontemporal: output is written once and never re-read by this kernel.
  {
    v4f*       dst  = reinterpret_cast<v4f*>(out + (size_t)b * ACC_FLOATS);
    const v4f* srcv = reinterpret_cast<const v4f*>(smem);
    for (int i = tid; i < ACC_FLOATS / 4; i += TPB)
      __builtin_nontemporal_store(srcv[i], &dst[i]);
  }
}

extern "C" void kernel_launch(void* const* d_in, const int* in_sizes, int n_in,
                              void* d_out, int out_size, void* d_ws, size_t ws_size,
                              hipStream_t stream) {
  const float* x     = (const float*)d_in[0];   // [N, F] f32
  const int*   z     = (const int*)d_in[1];     // [N]   int32 in [1,100]
  const int*   batch = (const int*)d_in[2];     // [N]   sorted graph ids
  float*       out   = (float*)d_out;           // [B, MAX_Z*F] f32

  const int N = in_sizes[1];                    // element count of z == N
  const int B = out_size / (MAX_Z * F);         // number of graphs

  hipLaunchKernelGGL(anp_pool_kernel, dim3(B), dim3(TPB), 0, stream,
                     x, z, batch, out, N);
}